// GIN_3l_5025111736763
// MI455X (gfx1250) — compile-verified
//
#include <hip/hip_runtime.h>

#define N_NODES   50000
#define N_EDGES   800000
#define DIM       128
#define N_GRAPHS  64
#define N_CLASSES 2

typedef __attribute__((ext_vector_type(2))) float v2f;
typedef __attribute__((ext_vector_type(8))) float v8f;

// ---------------------------------------------------------------------------
// agg init: agg = (1+eps)*h with eps=0  ->  plain vectorized copy
// ---------------------------------------------------------------------------
__global__ void copy_f4(const float4* __restrict__ in, float4* __restrict__ out, int n4) {
    int i = blockIdx.x * blockDim.x + threadIdx.x;
    if (i < n4) out[i] = in[i];
}

// ---------------------------------------------------------------------------
// edge scatter-add: agg[dst] += h[src] for src != dst.
// One thread handles (edge, 4 channels); 32 threads cover a 128-wide row.
// All buffers are L2-resident (25.6MB << 192MB L2), atomics hit L2 units.
// ---------------------------------------------------------------------------
__global__ void edge_scatter(const float* __restrict__ h, const int* __restrict__ src,
                             const int* __restrict__ dst, float* __restrict__ agg) {
    unsigned tid = blockIdx.x * blockDim.x + threadIdx.x;
    int e = tid >> 5;
    if (e >= N_EDGES) return;
    int c = (tid & 31) << 2;
    int s = src[e], d = dst[e];
    if (s == d) return;                       // self-loops removed, re-added via copy
    const float4 v = *(const float4*)(h + (size_t)s * DIM + c);
    float* p = agg + (size_t)d * DIM + c;
    atomicAdd(p + 0, v.x);
    atomicAdd(p + 1, v.y);
    atomicAdd(p + 2, v.z);
    atomicAdd(p + 3, v.w);
}

// ---------------------------------------------------------------------------
// out = relu(A[rows x 128] @ W[128 x 128] + b), fp32 WMMA 16x16x4.
// One wave per 16x16 output tile; K fully unrolled -> 32 v_wmma per wave.
// A layout (16x4 f32): lane m=L&15 is the row, half=L>>4 selects K pair.
// B layout (4x16 f32): lane n=L&15 is the col, half selects K pair.
// C/D: VGPR r holds M=r (lanes 0-15) and M=r+8 (lanes 16-31), N=lane&15.
// ---------------------------------------------------------------------------
__global__ __launch_bounds__(256)
void gemm128_wmma(const float* __restrict__ A, const float* __restrict__ W,
                  const float* __restrict__ bias, float* __restrict__ out,
                  int nRowTiles, int doRelu) {
    int tile = blockIdx.x * (blockDim.x >> 5) + (threadIdx.x >> 5);
    if (tile >= nRowTiles * 8) return;        // wave-uniform: EXEC stays all-1s
    int rowTile = tile >> 3;
    int col0    = (tile & 7) << 4;
    int lane    = threadIdx.x & 31;
    int m       = lane & 15;
    int half    = lane >> 4;

    const float* arow = A + (size_t)(rowTile * 16 + m) * DIM;
    const float* wcol = W + col0 + m;

    v8f acc = {};
#pragma unroll
    for (int k = 0; k < DIM; k += 4) {
        int kk = k + 2 * half;
        v2f a = *(const v2f*)(arow + kk);                       // A[m][kk], A[m][kk+1]
        v2f b = { wcol[(size_t)kk * DIM], wcol[(size_t)(kk + 1) * DIM] };
        acc = __builtin_amdgcn_wmma_f32_16x16x4_f32(
            /*neg_a=*/false, a, /*neg_b=*/false, b,
            /*c_mod=*/(short)0, acc, /*reuse_a=*/false, /*reuse_b=*/false);
    }

    float bv = bias[col0 + m];
#pragma unroll
    for (int r = 0; r < 8; ++r) {
        float v = acc[r] + bv;
        if (doRelu) v = fmaxf(v, 0.0f);
        out[(size_t)(rowTile * 16 + r + 8 * half) * DIM + col0 + m] = v;
    }
}

// ---------------------------------------------------------------------------
// pooling
// ---------------------------------------------------------------------------
__global__ void zero_f(float* __restrict__ p, int n) {
    int i = blockIdx.x * blockDim.x + threadIdx.x;
    if (i < n) p[i] = 0.0f;
}

__global__ void pool_accum(const float* __restrict__ h, const int* __restrict__ batch,
                           float* __restrict__ sums, float* __restrict__ cnts) {
    int n = blockIdx.x;        // one block per node
    int c = threadIdx.x;       // 128 channels
    int g = batch[n];
    atomicAdd(&sums[g * DIM + c], h[(size_t)n * DIM + c]);
    if (c == 0) atomicAdd(&cnts[g], 1.0f);
}

__global__ void pool_div(const float* __restrict__ sums, const float* __restrict__ cnts,
                         float* __restrict__ pooled) {
    int g = blockIdx.x, c = threadIdx.x;
    pooled[g * DIM + c] = sums[g * DIM + c] / fmaxf(cnts[g], 1.0f);
}

// ---------------------------------------------------------------------------
// final head: out[64 x 2] = pooled2 @ Wf2 + bf2   (tiny: one block)
// ---------------------------------------------------------------------------
__global__ void ffn_out(const float* __restrict__ pooled2, const float* __restrict__ Wf2,
                        const float* __restrict__ bf2, float* __restrict__ out) {
    int t   = threadIdx.x;     // 128 threads = 64 graphs x 2 classes
    int g   = t >> 1;
    int cls = t & 1;
    float s = bf2[cls];
#pragma unroll 8
    for (int k = 0; k < DIM; ++k)
        s += pooled2[g * DIM + k] * Wf2[k * N_CLASSES + cls];
    out[g * N_CLASSES + cls] = s;
}

// ---------------------------------------------------------------------------
extern "C" void kernel_launch(void* const* d_in, const int* in_sizes, int n_in,
                              void* d_out, int out_size, void* d_ws, size_t ws_size,
                              hipStream_t stream) {
    (void)in_sizes; (void)n_in; (void)out_size; (void)ws_size;

    const float* x     = (const float*)d_in[0];
    const int*   ei    = (const int*)d_in[1];
    const int*   src   = ei;
    const int*   dst   = ei + N_EDGES;
    const int*   batch = (const int*)d_in[2];
    const float* W[6]  = { (const float*)d_in[3],  (const float*)d_in[5],
                           (const float*)d_in[7],  (const float*)d_in[9],
                           (const float*)d_in[11], (const float*)d_in[13] };
    const float* b[6]  = { (const float*)d_in[4],  (const float*)d_in[6],
                           (const float*)d_in[8],  (const float*)d_in[10],
                           (const float*)d_in[12], (const float*)d_in[14] };
    const float* Wf1 = (const float*)d_in[15];
    const float* bf1 = (const float*)d_in[16];
    const float* Wf2 = (const float*)d_in[17];
    const float* bf2 = (const float*)d_in[18];
    float* out = (float*)d_out;

    const size_t NF = (size_t)N_NODES * DIM;   // 6.4M floats
    float* ws      = (float*)d_ws;
    float* bufA    = ws;                       // 25.6 MB
    float* bufB    = ws + NF;                  // 25.6 MB
    float* sums    = ws + 2 * NF;              // 64*128
    float* cnts    = sums + N_GRAPHS * DIM;    // 64
    float* pooled  = cnts + N_GRAPHS;          // 64*128
    float* pooled2 = pooled + N_GRAPHS * DIM;  // 64*128

    const int n4        = (int)(NF / 4);
    const int copyGrid  = (n4 + 255) / 256;
    const int scatGrid  = (int)(((size_t)N_EDGES * 32 + 255) / 256);
    const int rowTiles  = N_NODES / 16;        // 3125
    const int gemmGrid  = (rowTiles * 8) / 8;  // 8 waves/block of 256 threads

    // ---- layer 0: hin = x ----
    copy_f4<<<copyGrid, 256, 0, stream>>>((const float4*)x, (float4*)bufA, n4);
    edge_scatter<<<scatGrid, 256, 0, stream>>>(x, src, dst, bufA);
    gemm128_wmma<<<gemmGrid, 256, 0, stream>>>(bufA, W[0], b[0], bufB, rowTiles, 1);
    gemm128_wmma<<<gemmGrid, 256, 0, stream>>>(bufB, W[1], b[1], bufA, rowTiles, 1);
    // ---- layer 1: hin = bufA ----
    copy_f4<<<copyGrid, 256, 0, stream>>>((const float4*)bufA, (float4*)bufB, n4);
    edge_scatter<<<scatGrid, 256, 0, stream>>>(bufA, src, dst, bufB);
    gemm128_wmma<<<gemmGrid, 256, 0, stream>>>(bufB, W[2], b[2], bufA, rowTiles, 1);
    gemm128_wmma<<<gemmGrid, 256, 0, stream>>>(bufA, W[3], b[3], bufB, rowTiles, 1);
    // ---- layer 2: hin = bufB ----
    copy_f4<<<copyGrid, 256, 0, stream>>>((const float4*)bufB, (float4*)bufA, n4);
    edge_scatter<<<scatGrid, 256, 0, stream>>>(bufB, src, dst, bufA);
    gemm128_wmma<<<gemmGrid, 256, 0, stream>>>(bufA, W[4], b[4], bufB, rowTiles, 1);
    gemm128_wmma<<<gemmGrid, 256, 0, stream>>>(bufB, W[5], b[5], bufA, rowTiles, 1);
    // final node features in bufA

    // ---- global mean pool ----
    int nz = N_GRAPHS * DIM + N_GRAPHS;        // sums + cnts are contiguous
    zero_f<<<(nz + 255) / 256, 256, 0, stream>>>(sums, nz);
    pool_accum<<<N_NODES, DIM, 0, stream>>>(bufA, batch, sums, cnts);
    pool_div<<<N_GRAPHS, DIM, 0, stream>>>(sums, cnts, pooled);

    // ---- FFN ----
    gemm128_wmma<<<(N_GRAPHS / 16 * 8) / 8, 256, 0, stream>>>(pooled, Wf1, bf1, pooled2,
                                                              N_GRAPHS / 16, 1);
    ffn_out<<<1, 128, 0, stream>>>(pooled2, Wf2, bf2, out);
}